// SemiCRFDecoder_2087354106214
// MI455X (gfx1250) — compile-verified
//
#include <hip/hip_runtime.h>
#include <hip/hip_bf16.h>
#include <math.h>

// Problem constants (from reference)
#define B_ 4
#define T_ 320
#define L_ 96
#define E_ 64
#define T1_ (T_ + 1)
#define EPS_ 1e-8f
#define PCW_ 0.1f
#define ADVW_ 0.001f

typedef __attribute__((ext_vector_type(2))) float v2f;
typedef __attribute__((ext_vector_type(8))) float v8f;

// ---------------------------------------------------------------------------
// Kernel 1: per (c, b, l) — logits, softmax weights, prefix sums.
//   c = 0: cos(active, h)        c = 1: cos(active, 1-h)    c = 2: cos(1-active, h)
// Writes:
//   cwv[((c*B+b)*L+l)][t][e]  : prefix sums of w[t]*a_mod[t][e], t=0..T (row 0 = 0)
//   cd [((c*B+b)*L+l)][t]     : prefix sums of w[t]*d_raw[t]          (entry 0 = 0)
//   cad[b][t][l] (c==0 only)  : prefix sums of d_raw[t] / sum(h)      (entry 0 = 0)
//   hnorm[(c*B+b)*L+l]        : ||q_c||
// ---------------------------------------------------------------------------
__global__ __launch_bounds__(T_) void segscore_precompute(
    const float* __restrict__ act, const float* __restrict__ har,
    float* __restrict__ cwv, float* __restrict__ cd,
    float* __restrict__ cad, float* __restrict__ hnorm)
{
    const int c   = blockIdx.x / (B_ * L_);
    const int rem = blockIdx.x % (B_ * L_);
    const int b   = rem / L_;
    const int l   = rem % L_;
    const int t   = threadIdx.x;          // 0..T_-1

    __shared__ float h_lds[E_];
    __shared__ float red[T_];
    __shared__ float w_lds[T_];
    __shared__ float sm[2];               // [0]=max, [1]=1/hsum (c==0)

    if (t < E_) {
        float hv = har[(size_t)(b * L_ + l) * E_ + t];
        if (c == 1) hv = 1.0f - hv;
        h_lds[t] = hv;
    }
    __syncthreads();

    if (t == 0) {
        float s2 = 0.f, s1 = 0.f;
        for (int e = 0; e < E_; ++e) { s2 += h_lds[e] * h_lds[e]; s1 += h_lds[e]; }
        hnorm[(size_t)(c * B_ + b) * L_ + l] = sqrtf(s2);
        if (c == 0) sm[1] = 1.0f / s1;    // h is raw for c==0
    }
    __syncthreads();

    // raw dot d[t] = a_mod[t] . h_mod
    const float* arow = act + (size_t)(b * T_ + t) * E_;
    float dr = 0.f;
    for (int e = 0; e < E_; ++e) {
        float av = arow[e];
        if (c == 2) av = 1.0f - av;
        dr += av * h_lds[e];
    }
    const float S = dr * 0.125f;          // 1/sqrt(64)

    // global max over t
    red[t] = S;
    __syncthreads();
    if (t < 32) { float mx = red[t]; for (int j = t + 32; j < T_; j += 32) mx = fmaxf(mx, red[j]); red[t] = mx; }
    __syncthreads();
    if (t == 0) { float mx = red[0]; for (int j = 1; j < 32; ++j) mx = fmaxf(mx, red[j]); sm[0] = mx; }
    __syncthreads();

    const float w = expf(S - sm[0]);
    w_lds[t] = w;

    // inclusive scan of w*dr -> Cd
    float* cdp = cd + (size_t)((c * B_ + b) * L_ + l) * T1_;
    red[t] = w * dr;
    __syncthreads();
    for (int off = 1; off < T_; off <<= 1) {
        float x = red[t];
        float y = (t >= off) ? red[t - off] : 0.f;
        __syncthreads();
        red[t] = x + y;
        __syncthreads();
    }
    if (t == 0) cdp[0] = 0.f;
    cdp[t + 1] = red[t];

    if (c == 0) {   // uniform branch (c from blockIdx) — barriers safe
        __syncthreads();
        red[t] = dr;
        __syncthreads();
        for (int off = 1; off < T_; off <<= 1) {
            float x = red[t];
            float y = (t >= off) ? red[t - off] : 0.f;
            __syncthreads();
            red[t] = x + y;
            __syncthreads();
        }
        if (t == 0) cad[(size_t)(b * T1_ + 0) * L_ + l] = 0.f;
        cad[(size_t)(b * T1_ + t + 1) * L_ + l] = red[t] * sm[1];
    }
    __syncthreads();

    // column-wise running sums Cwv[t+1][e] = sum_{tt<=t} w[tt]*a_mod[tt][e]
    float* cwp = cwv + (size_t)((c * B_ + b) * L_ + l) * T1_ * E_;
    if (t < E_) {
        const int e = t;
        float acc = 0.f;
        cwp[e] = 0.f;                     // row 0
        for (int tt = 0; tt < T_; ++tt) {
            float av = act[(size_t)(b * T_ + tt) * E_ + e];
            if (c == 2) av = 1.0f - av;
            acc += w_lds[tt] * av;
            cwp[(size_t)(tt + 1) * E_ + e] = acc;
        }
    }
}

// ---------------------------------------------------------------------------
// Kernel 2: dg[i] = ||Cwv row i||^2, i over 3*B*L*(T+1) rows.
// ---------------------------------------------------------------------------
__global__ void segscore_dg(const float* __restrict__ cwv, float* __restrict__ dg)
{
    const int i = blockIdx.x * blockDim.x + threadIdx.x;
    if (i >= 3 * B_ * L_ * T1_) return;
    const float* p = cwv + (size_t)i * E_;
    float s = 0.f;
    for (int e = 0; e < E_; e += 4) {
        float4 v = *(const float4*)(p + e);
        s += v.x * v.x + v.y * v.y + v.z * v.z + v.w * v.w;
    }
    dg[i] = s;
}

// ---------------------------------------------------------------------------
// Kernel 3: WMMA Gram tiles + score assembly.
// grid = (s_tiles*e_tiles, L/16, B), block = 512 (16 waves, wave w -> l = lg*16+w)
// Each wave: per variant c, G16x16 = A[16x64] x B^T[64x16] via 16x v_wmma_f32_16x16x4_f32.
// All 32 operand b64 loads are issued as one clause, then the 16 WMMAs run as
// two independent accumulation chains (even/odd k) merged at the end.
// A layout: lane = M (row s0+lane&15), half selects K pair; v2f = {K, K+1}.
// B layout mirrored with lane = N (row e0+1+lane&15).
// ---------------------------------------------------------------------------
__global__ __launch_bounds__(512) void segscore_main(
    const float* __restrict__ cwv, const float* __restrict__ cd,
    const float* __restrict__ dg,  const float* __restrict__ cad,
    const float* __restrict__ hnorm, float* __restrict__ out)
{
    const int et   = blockIdx.x % (T_ / 16);
    const int st   = blockIdx.x / (T_ / 16);
    const int lg   = blockIdx.y;
    const int b    = blockIdx.z;
    const int wave = threadIdx.x >> 5;
    const int lane = threadIdx.x & 31;
    const int l    = lg * 16 + wave;
    const int s0   = st * 16;
    const int e0   = et * 16;
    const int half = lane >> 4;
    const int lid  = lane & 15;

    __shared__ float tile[16][16][16];    // [M][N][wave]

    const int rowA = s0 + lid;            // A-matrix row (segment start s)
    const int rowB = e0 + 1 + lid;        // B-matrix row (prefix index e+1)
    const int n    = lid;                 // N coordinate of this lane's C entries
    const int eg   = e0 + n;              // segment end e
    const int col  = eg + 1;

    float score[8];
#pragma unroll
    for (int r = 0; r < 8; ++r) score[r] = 0.f;

    for (int c = 0; c < 3; ++c) {
        const size_t pl = (size_t)((c * B_ + b) * L_ + l);
        const float* P   = cwv + pl * T1_ * E_;
        const float* Cdp = cd + pl * T1_;
        const float* Dgp = dg + pl * T1_;
        const float  hn  = hnorm[pl];
        const float  wc  = (c == 0) ? 1.0f : -PCW_;   // immediate, no const-mem load

        const float* pa = P + (size_t)rowA * E_ + 2 * half;
        const float* pb = P + (size_t)rowB * E_ + 2 * half;

        // Preload all operands (compiler emits these as one load clause).
        v2f areg[16], breg[16];
#pragma unroll
        for (int k = 0; k < 16; ++k) {
            areg[k] = *(const v2f*)(pa + 4 * k);
            breg[k] = *(const v2f*)(pb + 4 * k);
        }

        // Two independent WMMA accumulation chains over K = 64.
        v8f acc0 = {}, acc1 = {};
#pragma unroll
        for (int k = 0; k < 16; k += 2) {
            acc0 = __builtin_amdgcn_wmma_f32_16x16x4_f32(
                false, areg[k],     false, breg[k],     (short)0, acc0, false, false);
            acc1 = __builtin_amdgcn_wmma_f32_16x16x4_f32(
                false, areg[k + 1], false, breg[k + 1], (short)0, acc1, false, false);
        }

        const float cd_col = Cdp[col];
        const float dg_col = Dgp[col];
        const float hden   = fmaxf(hn, EPS_);
#pragma unroll
        for (int r = 0; r < 8; ++r) {
            const int s = s0 + 8 * half + r;
            const float g   = acc0[r] + acc1[r];
            const float num = cd_col - Cdp[s];
            const float nsq = dg_col + Dgp[s] - 2.0f * g;
            const float na  = sqrtf(fmaxf(nsq, 0.f));
            score[r] += wc * num / (fmaxf(na, EPS_) * hden);
        }
    }

    // adversarial term + validity mask
    const float ca_col = cad[(size_t)(b * T1_ + col) * L_ + l];
#pragma unroll
    for (int r = 0; r < 8; ++r) {
        const int s = s0 + 8 * half + r;
        const float seg = ca_col - cad[(size_t)(b * T1_ + s) * L_ + l];
        const float len = fmaxf((float)(eg - s + 1), 1.0f);
        float v = score[r] - ADVW_ * seg / len;
        if (s > eg) v = 0.f;
        tile[8 * half + r][n][wave] = v;
    }
    __syncthreads();

    // coalesced store: 4096 floats, 8 consecutive (in l) per thread
    const int l0 = lg * 16;
#pragma unroll
    for (int j = 0; j < 8; ++j) {
        const int f  = threadIdx.x * 8 + j;
        const int li = f & 15;
        const int nn = (f >> 4) & 15;
        const int mm = f >> 8;
        out[(((size_t)b * T_ + (s0 + mm)) * T_ + (e0 + nn)) * L_ + (l0 + li)] =
            tile[mm][nn][li];
    }
}

// ---------------------------------------------------------------------------
extern "C" void kernel_launch(void* const* d_in, const int* in_sizes, int n_in,
                              void* d_out, int out_size, void* d_ws, size_t ws_size,
                              hipStream_t stream)
{
    const float* act = (const float*)d_in[0];   // active_pc  [B,T,E]
    // d_in[1] (inactive_pc) is unused by the reference math
    const float* har = (const float*)d_in[2];   // harmony_pc [B,L,E]

    // Workspace layout (floats); total ~98 MB, fits in 192 MB L2.
    float* ws   = (float*)d_ws;
    float* cwv  = ws;                                           // 3*B*L*(T+1)*E
    float* cd_  = cwv + (size_t)3 * B_ * L_ * T1_ * E_;         // 3*B*L*(T+1)
    float* dg_  = cd_ + (size_t)3 * B_ * L_ * T1_;              // 3*B*L*(T+1)
    float* cad_ = dg_ + (size_t)3 * B_ * L_ * T1_;              // B*(T+1)*L
    float* hn_  = cad_ + (size_t)B_ * T1_ * L_;                 // 3*B*L

    segscore_precompute<<<3 * B_ * L_, T_, 0, stream>>>(act, har, cwv, cd_, cad_, hn_);

    const int ndg = 3 * B_ * L_ * T1_;
    segscore_dg<<<(ndg + 255) / 256, 256, 0, stream>>>(cwv, dg_);

    dim3 grid((T_ / 16) * (T_ / 16), L_ / 16, B_);
    segscore_main<<<grid, 512, 0, stream>>>(cwv, cd_, dg_, cad_, hn_, (float*)d_out);
}